// SDGraphEncoder_39152921870560
// MI455X (gfx1250) — compile-verified
//
#include <hip/hip_runtime.h>
#include <hip/hip_bf16.h>

typedef float v2f __attribute__((ext_vector_type(2)));
typedef float v8f __attribute__((ext_vector_type(8)));

#define BSZ   4
#define CSP   64
#define CDN   64
#define NSTK  32
#define NPNT  128
#define KNN   20
#define NN    4096      // NSTK*NPNT
#define UU    128
#define SPOUT 128
#define DNOUT 128
#define EPSV  1e-5f

__device__ __forceinline__ float leakyf(float x) { return x > 0.f ? x : 0.2f * x; }

// ---------------------------------------------------------------------------
// Kernel 1: DenseToSparse  conv(1,3,valid) + BN2d + leaky + max over points
// one block per out-channel (64 blocks, 128 threads)
// ---------------------------------------------------------------------------
__global__ void k_d2s(const float* __restrict__ dense, const float* __restrict__ W,
                      const float* __restrict__ bias, const float* __restrict__ g,
                      const float* __restrict__ be, float* __restrict__ convbuf,
                      float* __restrict__ d2smax) {
    const int c = blockIdx.x, tid = threadIdx.x;
    const int PW = NPNT - 2;            // 126
    const int E  = BSZ * NSTK * PW;     // 16128
    __shared__ float s1[128], s2[128];
    __shared__ float smean, sscale, sbeta;
    float a1 = 0.f, a2 = 0.f;
    float* cb = convbuf + (size_t)c * E;
    const float* wc = W + c * CDN * 3;
    for (int e = tid; e < E; e += 128) {
        int b = e / (NSTK * PW);
        int rem = e - b * (NSTK * PW);
        int s = rem / PW, p = rem - s * PW;
        float v = bias[c];
        const float* dbase = dense + (((size_t)b * CDN) * NSTK + s) * NPNT + p;
        for (int ci = 0; ci < CDN; ++ci) {
            const float* dp = dbase + (size_t)ci * NSTK * NPNT;
            const float* wp = wc + ci * 3;
            v += dp[0] * wp[0] + dp[1] * wp[1] + dp[2] * wp[2];
        }
        cb[e] = v; a1 += v; a2 += v * v;
    }
    s1[tid] = a1; s2[tid] = a2; __syncthreads();
    for (int s = 64; s > 0; s >>= 1) {
        if (tid < s) { s1[tid] += s1[tid + s]; s2[tid] += s2[tid + s]; }
        __syncthreads();
    }
    if (tid == 0) {
        float mean = s1[0] / (float)E;
        float var  = s2[0] / (float)E - mean * mean;
        smean = mean; sscale = g[c] * rsqrtf(var + EPSV); sbeta = be[c];
    }
    __syncthreads();
    // phase 2: max over point dim, one (b,s) pair per thread
    int b = tid >> 5, s = tid & 31;
    const float* row = cb + (size_t)(b * NSTK + s) * PW;
    float mx = -3.0e38f;
    for (int p = 0; p < PW; ++p) {
        float y = leakyf((row[p] - smean) * sscale + sbeta);
        mx = fmaxf(mx, y);
    }
    d2smax[(b * CDN + c) * NSTK + s] = mx;
}

// ---------------------------------------------------------------------------
// Kernel 2: sparse update  zs = leaky(BN1d(W_sp @ union_sparse + b))
// one block per out-channel (128 blocks, 128 threads = 128 columns)
// ---------------------------------------------------------------------------
__global__ void k_zs(const float* __restrict__ sparse, const float* __restrict__ d2smax,
                     const float* __restrict__ Wsp, const float* __restrict__ bsp,
                     const float* __restrict__ gsp, const float* __restrict__ besp,
                     float* __restrict__ outzs) {
    const int o = blockIdx.x, tid = threadIdx.x;
    const int b = tid >> 5, s = tid & 31;
    __shared__ float s1[128], s2[128];
    __shared__ float smean, sscale, sbeta;
    const float* w = Wsp + o * UU;
    float v = bsp[o];
    for (int c = 0; c < CSP; ++c) v += w[c] * sparse[(b * CSP + c) * NSTK + s];
    for (int c = 0; c < CDN; ++c) v += w[CSP + c] * d2smax[(b * CDN + c) * NSTK + s];
    s1[tid] = v; s2[tid] = v * v; __syncthreads();
    for (int st = 64; st > 0; st >>= 1) {
        if (tid < st) { s1[tid] += s1[tid + st]; s2[tid] += s2[tid + st]; }
        __syncthreads();
    }
    if (tid == 0) {
        float mean = s1[0] / 128.f;
        float var  = s2[0] / 128.f - mean * mean;
        smean = mean; sscale = gsp[o] * rsqrtf(var + EPSV); sbeta = besp[o];
    }
    __syncthreads();
    outzs[(b * SPOUT + o) * NSTK + s] = leakyf((v - smean) * sscale + sbeta);
}

// ---------------------------------------------------------------------------
// Kernel 3: build xt[b][n][c] (row-major, K contiguous) and sq[b][n]
// grid (NN, BSZ), 128 threads (one per channel)
// ---------------------------------------------------------------------------
__global__ void k_build_xt(const float* __restrict__ dense, const float* __restrict__ sparse,
                           float* __restrict__ xt, float* __restrict__ sq) {
    const int n = blockIdx.x, b = blockIdx.y, c = threadIdx.x;
    const int s = n >> 7, p = n & 127;
    float v = (c < CDN) ? dense[(((size_t)b * CDN + c) * NSTK + s) * NPNT + p]
                        : sparse[(b * CSP + (c - CDN)) * NSTK + s];
    xt[(((size_t)b * NN) + n) * UU + c] = v;
    __shared__ float sh[128];
    sh[c] = v * v; __syncthreads();
    for (int st = 64; st > 0; st >>= 1) {
        if (c < st) sh[c] += sh[c + st];
        __syncthreads();
    }
    if (c == 0) sq[(size_t)b * NN + n] = sh[0];
}

// ---------------------------------------------------------------------------
// Kernel 4: G = X @ X^T  (per batch), f32 WMMA 16x16x4, SYMMETRIC:
// only tiles with colBase >= rowBase are computed (~half the WMMA work);
// mirrored half is written via an LDS-transposed, coalesced store.
// WMMA accumulates the same products in the same k-order for (n,m) and (m,n),
// so direct and mirrored values are bitwise identical -> deterministic.
// block = 256 threads = 8 waves; wave computes a 16x64 strip (4 accumulators)
// grid: (NN/64, NN/128)
// ---------------------------------------------------------------------------
__global__ void k_gemm_xxt(const float* __restrict__ X, float* __restrict__ G) {
    const int rowBase = blockIdx.y * 128;
    const int colBase = blockIdx.x * 64;
    if (colBase < rowBase) return;      // symmetric skip (uniform per block)

    __shared__ float tile[128 * 65];    // 65-float pitch: conflict-light transpose
    const int lane = threadIdx.x & 31;
    const int wave = threadIdx.x >> 5;
    const int lr = lane & 15, half = lane >> 4;
    const int wrow = rowBase + wave * 16;

    const float* arow  = X + (size_t)(wrow + lr) * UU + 2 * half;
    const float* brow0 = X + (size_t)(colBase +  0 + lr) * UU + 2 * half;
    const float* brow1 = X + (size_t)(colBase + 16 + lr) * UU + 2 * half;
    const float* brow2 = X + (size_t)(colBase + 32 + lr) * UU + 2 * half;
    const float* brow3 = X + (size_t)(colBase + 48 + lr) * UU + 2 * half;

    v8f acc0 = {0,0,0,0,0,0,0,0}, acc1 = acc0, acc2 = acc0, acc3 = acc0;
    for (int kk = 0; kk < UU; kk += 4) {
        v2f a  = *reinterpret_cast<const v2f*>(arow + kk);
        v2f b0 = *reinterpret_cast<const v2f*>(brow0 + kk);
        v2f b1 = *reinterpret_cast<const v2f*>(brow1 + kk);
        v2f b2 = *reinterpret_cast<const v2f*>(brow2 + kk);
        v2f b3 = *reinterpret_cast<const v2f*>(brow3 + kk);
        acc0 = __builtin_amdgcn_wmma_f32_16x16x4_f32(false, a, false, b0, (short)0, acc0, false, false);
        acc1 = __builtin_amdgcn_wmma_f32_16x16x4_f32(false, a, false, b1, (short)0, acc1, false, false);
        acc2 = __builtin_amdgcn_wmma_f32_16x16x4_f32(false, a, false, b2, (short)0, acc2, false, false);
        acc3 = __builtin_amdgcn_wmma_f32_16x16x4_f32(false, a, false, b3, (short)0, acc3, false, false);
    }
    #pragma unroll
    for (int r = 0; r < 8; ++r) {
        const int ri = r + 8 * half;                 // 0..15 within wave tile
        const size_t grow = (size_t)(wrow + ri) * NN;
        G[grow + colBase +  0 + lr] = acc0[r];
        G[grow + colBase + 16 + lr] = acc1[r];
        G[grow + colBase + 32 + lr] = acc2[r];
        G[grow + colBase + 48 + lr] = acc3[r];
        const int li = (wave * 16 + ri) * 65;
        tile[li +  0 + lr] = acc0[r];
        tile[li + 16 + lr] = acc1[r];
        tile[li + 32 + lr] = acc2[r];
        tile[li + 48 + lr] = acc3[r];
    }
    __syncthreads();
    // mirror store: G[colBase+j][rowBase+i] = tile[i][j]
    // thread t: j = t>>2 (row of mirror), i0 = (t&3)*32 -> 32 consecutive floats
    const int t = threadIdx.x;
    const int j = t >> 2;
    const int i0 = (t & 3) * 32;
    float* gm = G + (size_t)(colBase + j) * NN + rowBase + i0;
    #pragma unroll
    for (int ii = 0; ii < 32; ++ii) gm[ii] = tile[(i0 + ii) * 65 + j];
}

// ---------------------------------------------------------------------------
// Kernel 5: top-(K+1) smallest dist per row; key = sq[m] - 2*G[n][m]
// (sq[n] is a row-constant so ordering is unchanged). Drop the smallest (self).
// one block per row n, 128 threads, per-thread sorted lists in LDS + k-way merge.
// ---------------------------------------------------------------------------
__global__ void k_topk(const float* __restrict__ G, const float* __restrict__ sqb,
                       int* __restrict__ idxb) {
    const int n = blockIdx.x, tid = threadIdx.x;
    const int L = KNN + 1; // 21
    __shared__ float lv[128 * 21];
    __shared__ int   li[128 * 21];
    __shared__ int   ptr[128];
    __shared__ float rv[128];
    __shared__ int   rm[128];
    __shared__ int   rl[128];

    const float* Grow = G + (size_t)n * NN;
    const int base = tid * L;
    int cnt = 0;
    for (int m = tid; m < NN; m += 128) {
        float key = sqb[m] - 2.0f * Grow[m];
        if (cnt == L) {
            float lvl = lv[base + L - 1]; int lil = li[base + L - 1];
            if (!(key < lvl || (key == lvl && m < lil))) continue;
            int pos = L - 1;
            while (pos > 0) {
                float pv = lv[base + pos - 1]; int pi = li[base + pos - 1];
                if (key < pv || (key == pv && m < pi)) {
                    lv[base + pos] = pv; li[base + pos] = pi; --pos;
                } else break;
            }
            lv[base + pos] = key; li[base + pos] = m;
        } else {
            int pos = cnt;
            while (pos > 0) {
                float pv = lv[base + pos - 1]; int pi = li[base + pos - 1];
                if (key < pv || (key == pv && m < pi)) {
                    lv[base + pos] = pv; li[base + pos] = pi; --pos;
                } else break;
            }
            lv[base + pos] = key; li[base + pos] = m;
            ++cnt;
        }
    }
    ptr[tid] = 0;
    __syncthreads();
    // 128-way merge: 21 rounds of parallel lexicographic argmin over list heads
    for (int j = 0; j < L; ++j) {
        int p = ptr[tid];
        if (p < L) { rv[tid] = lv[base + p]; rm[tid] = li[base + p]; }
        else       { rv[tid] = 3.0e38f;      rm[tid] = 0x7fffffff;   }
        rl[tid] = tid;
        __syncthreads();
        for (int st = 64; st > 0; st >>= 1) {
            if (tid < st) {
                float v2 = rv[tid + st]; int m2 = rm[tid + st];
                if (v2 < rv[tid] || (v2 == rv[tid] && m2 < rm[tid])) {
                    rv[tid] = v2; rm[tid] = m2; rl[tid] = rl[tid + st];
                }
            }
            __syncthreads();
        }
        if (tid == 0 && j > 0) idxb[(size_t)n * KNN + (j - 1)] = rm[0];
        if (tid == rl[0]) ptr[tid] = ptr[tid] + 1;
        __syncthreads();
    }
}

// ---------------------------------------------------------------------------
// Kernel 6: [P;Q] = [W1;W2] (256x128) @ X^T (128x4096), f32 WMMA, per batch
// W1 = W_dn[:, :128], W2 = W_dn[:, 128:], R[b][oc][n]
// grid (NN/64, 256/128, BSZ), 256 threads
// ---------------------------------------------------------------------------
__global__ void k_gemm_pq(const float* __restrict__ Wdn, const float* __restrict__ X,
                          float* __restrict__ R) {
    const int lane = threadIdx.x & 31;
    const int wave = threadIdx.x >> 5;
    const int lr = lane & 15, half = lane >> 4;
    const int rowBase = blockIdx.y * 128 + wave * 16;
    const int colBase = blockIdx.x * 64;
    const int b = blockIdx.z;
    const float* Xb = X + (size_t)b * NN * UU;
    float* Rb = R + (size_t)b * 2 * DNOUT * NN;

    const int r = rowBase + lr;
    const float* wrow = (r < DNOUT) ? (Wdn + (size_t)r * (2 * UU))
                                    : (Wdn + (size_t)(r - DNOUT) * (2 * UU) + UU);
    wrow += 2 * half;
    const float* brow0 = Xb + (size_t)(colBase +  0 + lr) * UU + 2 * half;
    const float* brow1 = Xb + (size_t)(colBase + 16 + lr) * UU + 2 * half;
    const float* brow2 = Xb + (size_t)(colBase + 32 + lr) * UU + 2 * half;
    const float* brow3 = Xb + (size_t)(colBase + 48 + lr) * UU + 2 * half;

    v8f acc0 = {0,0,0,0,0,0,0,0}, acc1 = acc0, acc2 = acc0, acc3 = acc0;
    for (int kk = 0; kk < UU; kk += 4) {
        v2f a  = *reinterpret_cast<const v2f*>(wrow + kk);
        v2f b0 = *reinterpret_cast<const v2f*>(brow0 + kk);
        v2f b1 = *reinterpret_cast<const v2f*>(brow1 + kk);
        v2f b2 = *reinterpret_cast<const v2f*>(brow2 + kk);
        v2f b3 = *reinterpret_cast<const v2f*>(brow3 + kk);
        acc0 = __builtin_amdgcn_wmma_f32_16x16x4_f32(false, a, false, b0, (short)0, acc0, false, false);
        acc1 = __builtin_amdgcn_wmma_f32_16x16x4_f32(false, a, false, b1, (short)0, acc1, false, false);
        acc2 = __builtin_amdgcn_wmma_f32_16x16x4_f32(false, a, false, b2, (short)0, acc2, false, false);
        acc3 = __builtin_amdgcn_wmma_f32_16x16x4_f32(false, a, false, b3, (short)0, acc3, false, false);
    }
    #pragma unroll
    for (int rr = 0; rr < 8; ++rr) {
        const size_t orow = (size_t)(rowBase + rr + 8 * half) * NN;
        Rb[orow + colBase +  0 + lr] = acc0[rr];
        Rb[orow + colBase + 16 + lr] = acc1[rr];
        Rb[orow + colBase + 32 + lr] = acc2[rr];
        Rb[orow + colBase + 48 + lr] = acc3[rr];
    }
}

// ---------------------------------------------------------------------------
// Kernel 7: edge values via gather: v = P[idx] - P[n] + Q[n] + bias
// per (o,b) block: max/min over k per n, and partial sum/sumsq for BN stats
// grid (DNOUT, BSZ), 256 threads
// ---------------------------------------------------------------------------
__global__ void k_edge(const float* __restrict__ R, const int* __restrict__ idx,
                       const float* __restrict__ bdn, float* __restrict__ maxv,
                       float* __restrict__ minv, float* __restrict__ partial) {
    const int o = blockIdx.x, b = blockIdx.y, tid = threadIdx.x;
    __shared__ float s1[256], s2[256];
    const float* P = R + (size_t)(b * 2 * DNOUT + o) * NN;
    const float* Q = P + (size_t)DNOUT * NN;
    const int* id = idx + (size_t)b * NN * KNN;
    const float bias = bdn[o];
    float a1 = 0.f, a2 = 0.f;
    for (int n = tid; n < NN; n += 256) {
        const float base = Q[n] - P[n] + bias;
        float mx = -3.0e38f, mn = 3.0e38f;
        const int* idn = id + (size_t)n * KNN;
        #pragma unroll
        for (int k = 0; k < KNN; ++k) {
            float v = P[idn[k]] + base;
            mx = fmaxf(mx, v); mn = fminf(mn, v);
            a1 += v; a2 += v * v;
        }
        maxv[((size_t)(b * DNOUT + o)) * NN + n] = mx;
        minv[((size_t)(b * DNOUT + o)) * NN + n] = mn;
    }
    s1[tid] = a1; s2[tid] = a2; __syncthreads();
    for (int st = 128; st > 0; st >>= 1) {
        if (tid < st) { s1[tid] += s1[tid + st]; s2[tid] += s2[tid + st]; }
        __syncthreads();
    }
    if (tid == 0) {
        partial[(o * BSZ + b) * 2 + 0] = s1[0];
        partial[(o * BSZ + b) * 2 + 1] = s2[0];
    }
}

// ---------------------------------------------------------------------------
// Kernel 8: finalize dense update: BN2d + leaky + (max over k via max/min select)
// grid (DNOUT, BSZ), 256 threads
// ---------------------------------------------------------------------------
__global__ void k_zd3(const float* __restrict__ maxv, const float* __restrict__ minv,
                      const float* __restrict__ partial, const float* __restrict__ gdn,
                      const float* __restrict__ bedn, float* __restrict__ zd3) {
    const int o = blockIdx.x, b = blockIdx.y, tid = threadIdx.x;
    float s1 = 0.f, s2 = 0.f;
    for (int bb = 0; bb < BSZ; ++bb) {   // deterministic cross-block sum
        s1 += partial[(o * BSZ + bb) * 2 + 0];
        s2 += partial[(o * BSZ + bb) * 2 + 1];
    }
    const float cnt = (float)(BSZ * NN * KNN);
    const float mean = s1 / cnt;
    const float var  = s2 / cnt - mean * mean;
    const float scale = gdn[o] * rsqrtf(var + EPSV);
    const float beta = bedn[o];
    const size_t off = (size_t)(b * DNOUT + o) * NN;
    for (int n = tid; n < NN; n += 256) {
        float sel = (scale >= 0.f) ? maxv[off + n] : minv[off + n];
        zd3[off + n] = leakyf((sel - mean) * scale + beta);
    }
}

// ---------------------------------------------------------------------------
// Kernel 9: DownSample conv(1,3,stride2,pad1) + BN2d + leaky
// one block per out-channel (128 blocks, 256 threads)
// ---------------------------------------------------------------------------
__global__ void k_ds(const float* __restrict__ zd3, const float* __restrict__ Wds,
                     const float* __restrict__ bds, const float* __restrict__ gds,
                     const float* __restrict__ beds, float* __restrict__ convds,
                     float* __restrict__ outzd) {
    const int o = blockIdx.x, tid = threadIdx.x;
    const int PW = NPNT / 2;                 // 64
    const int E = BSZ * NSTK * PW;           // 8192
    __shared__ float s1[256], s2[256];
    __shared__ float smean, sscale, sbeta;
    float a1 = 0.f, a2 = 0.f;
    float* cb = convds + (size_t)o * E;
    const float* wo = Wds + (size_t)o * DNOUT * 3;
    for (int e = tid; e < E; e += 256) {
        int b = e / (NSTK * PW);
        int rem = e - b * (NSTK * PW);
        int s = rem / PW, pp = rem - s * PW;
        int ip0 = 2 * pp - 1;
        float v = bds[o];
        for (int ci = 0; ci < DNOUT; ++ci) {
            const float* in = zd3 + (size_t)(b * DNOUT + ci) * NN + s * NPNT;
            const float* wp = wo + ci * 3;
            #pragma unroll
            for (int kw = 0; kw < 3; ++kw) {
                int ip = ip0 + kw;
                if (ip >= 0 && ip < NPNT) v += in[ip] * wp[kw];
            }
        }
        cb[e] = v; a1 += v; a2 += v * v;
    }
    s1[tid] = a1; s2[tid] = a2; __syncthreads();
    for (int st = 128; st > 0; st >>= 1) {
        if (tid < st) { s1[tid] += s1[tid + st]; s2[tid] += s2[tid + st]; }
        __syncthreads();
    }
    if (tid == 0) {
        float mean = s1[0] / (float)E;
        float var  = s2[0] / (float)E - mean * mean;
        smean = mean; sscale = gds[o] * rsqrtf(var + EPSV); sbeta = beds[o];
    }
    __syncthreads();
    for (int e = tid; e < E; e += 256) {
        int b = e / (NSTK * PW);
        int rem = e - b * (NSTK * PW);
        int s = rem / PW, pp = rem - s * PW;
        float y = leakyf((cb[e] - smean) * sscale + sbeta);
        outzd[((size_t)(b * DNOUT + o) * NSTK + s) * PW + pp] = y;
    }
}

// ---------------------------------------------------------------------------
extern "C" void kernel_launch(void* const* d_in, const int* in_sizes, int n_in,
                              void* d_out, int out_size, void* d_ws, size_t ws_size,
                              hipStream_t stream) {
    (void)in_sizes; (void)n_in; (void)out_size; (void)ws_size;
    const float* sparse = (const float*)d_in[0];
    const float* dense  = (const float*)d_in[1];
    const float* W_d2s  = (const float*)d_in[2];
    const float* b_d2s  = (const float*)d_in[3];
    const float* g_d2s  = (const float*)d_in[4];
    const float* be_d2s = (const float*)d_in[5];
    const float* W_sp   = (const float*)d_in[6];
    const float* b_sp   = (const float*)d_in[7];
    const float* g_sp   = (const float*)d_in[8];
    const float* be_sp  = (const float*)d_in[9];
    const float* W_dn   = (const float*)d_in[10];
    const float* b_dn   = (const float*)d_in[11];
    const float* g_dn   = (const float*)d_in[12];
    const float* be_dn  = (const float*)d_in[13];
    const float* W_ds   = (const float*)d_in[14];
    const float* b_ds   = (const float*)d_in[15];
    const float* g_ds   = (const float*)d_in[16];
    const float* be_ds  = (const float*)d_in[17];

    float* out = (float*)d_out;
    float* out_zs = out;                                   // [4,128,32]
    float* out_zd = out + (size_t)BSZ * SPOUT * NSTK;      // [4,128,32,64]

    // workspace layout (floats)
    float* base = (float*)d_ws;
    size_t off = 0;
    float* xt      = base + off; off += (size_t)BSZ * NN * UU;         // 2,097,152
    float* sq      = base + off; off += (size_t)BSZ * NN;              // 16,384
    float* d2smax  = base + off; off += (size_t)BSZ * CDN * NSTK;      // 8,192
    float* convbuf = base + off; off += (size_t)CDN * BSZ * NSTK * 126;// 1,032,192
    float* R       = base + off; off += (size_t)BSZ * 2 * DNOUT * NN;  // 4,194,304
    int*   idxbuf  = (int*)(base + off); off += (size_t)BSZ * NN * KNN;// 327,680
    float* maxv    = base + off; off += (size_t)BSZ * DNOUT * NN;      // 2,097,152
    float* minv    = base + off; off += (size_t)BSZ * DNOUT * NN;      // 2,097,152
    float* partial = base + off; off += (size_t)DNOUT * BSZ * 2;       // 1,024
    float* zd3     = base + off; off += (size_t)BSZ * DNOUT * NN;      // 2,097,152
    float* convds  = base + off; off += (size_t)DNOUT * BSZ * NSTK * 64;//1,048,576
    float* G       = base + off; off += (size_t)NN * NN;               // 16,777,216

    // small paths
    k_d2s<<<dim3(CDN), 128, 0, stream>>>(dense, W_d2s, b_d2s, g_d2s, be_d2s, convbuf, d2smax);
    k_zs<<<dim3(SPOUT), 128, 0, stream>>>(sparse, d2smax, W_sp, b_sp, g_sp, be_sp, out_zs);

    // build xt / sq
    k_build_xt<<<dim3(NN, BSZ), 128, 0, stream>>>(dense, sparse, xt, sq);

    // P/Q GEMM (WMMA) — all batches
    k_gemm_pq<<<dim3(NN / 64, 2 * DNOUT / 128, BSZ), 256, 0, stream>>>(W_dn, xt, R);

    // distance GEMM (symmetric WMMA) + top-K, per batch (G buffer reused)
    for (int b = 0; b < BSZ; ++b) {
        k_gemm_xxt<<<dim3(NN / 64, NN / 128), 256, 0, stream>>>(xt + (size_t)b * NN * UU, G);
        k_topk<<<dim3(NN), 128, 0, stream>>>(G, sq + (size_t)b * NN, idxbuf + (size_t)b * NN * KNN);
    }

    // edge gather + stats, finalize, downsample
    k_edge<<<dim3(DNOUT, BSZ), 256, 0, stream>>>(R, idxbuf, b_dn, maxv, minv, partial);
    k_zd3<<<dim3(DNOUT, BSZ), 256, 0, stream>>>(maxv, minv, partial, g_dn, be_dn, zd3);
    k_ds<<<dim3(DNOUT), 256, 0, stream>>>(zd3, W_ds, b_ds, g_ds, be_ds, convds, out_zd);
}